// MultiGraspToleranceLoss_5282809774825
// MI455X (gfx1250) — compile-verified
//
#include <hip/hip_runtime.h>
#include <math.h>

// ---------------------------------------------------------------------------
// MultiGraspToleranceLoss for MI455X (gfx1250, wave32)
//
// Memory-bound streaming problem (~107 MB read, ~4.6 us at 23.3 TB/s HBM).
// Deterministic reductions only (no float atomics). exp/segment-max fused to
// exp(-segment_min/k) by monotonicity. Final 1024x4 partial reduction done
// with V_WMMA_F32_16X16X4_F32 (A = ones -> column sums of B, layout-robust,
// exact f32 accumulation in C).
// ---------------------------------------------------------------------------

#define BS   256     // main/block-sum block size
#define SCAN 1024    // max blocks handled by the single-block scan

typedef __attribute__((ext_vector_type(2))) float v2f;
typedef __attribute__((ext_vector_type(8))) float v8f;

// ---- Kernel 1: per-block sums of gt_counts (deterministic tree) ------------
__global__ void k_block_sums(const int* __restrict__ counts,
                             int* __restrict__ blockSums, int n)
{
    __shared__ int s[BS];
    int tid = threadIdx.x;
    int i   = blockIdx.x * BS + tid;
    s[tid]  = (i < n) ? counts[i] : 0;
    __syncthreads();
    for (int st = BS / 2; st > 0; st >>= 1) {
        if (tid < st) s[tid] += s[tid + st];
        __syncthreads();
    }
    if (tid == 0) blockSums[blockIdx.x] = s[0];
}

// ---- Kernel 2: exclusive scan of block sums (single block, Hillis-Steele) --
__global__ void k_scan_blocks(const int* __restrict__ blockSums,
                              int* __restrict__ blockOffsets, int nb)
{
    __shared__ int s[SCAN];
    int tid = threadIdx.x;
    int v   = (tid < nb) ? blockSums[tid] : 0;
    s[tid]  = v;
    __syncthreads();
    for (int off = 1; off < SCAN; off <<= 1) {
        int t = (tid >= off) ? s[tid - off] : 0;
        __syncthreads();
        s[tid] += t;
        __syncthreads();
    }
    if (tid < nb) blockOffsets[tid] = s[tid] - v;   // exclusive
}

// ---- Kernel 3: main per-segment loss + deterministic block reduction -------
__global__ void k_main(const float* __restrict__ pred,
                       const float* __restrict__ gt,
                       const int*   __restrict__ counts,
                       const int*   __restrict__ blockOffsets,
                       float* __restrict__ partials, int n)
{
    __shared__ int   sscan[BS];
    __shared__ float sred[4][BS];

    int tid = threadIdx.x;
    int i   = blockIdx.x * BS + tid;
    int c   = (i < n) ? counts[i] : 0;

    // local exclusive scan of counts within the block
    sscan[tid] = c;
    __syncthreads();
    for (int off = 1; off < BS; off <<= 1) {
        int t = (tid >= off) ? sscan[tid - off] : 0;
        __syncthreads();
        sscan[tid] += t;
        __syncthreads();
    }
    int o = blockOffsets[blockIdx.x] + sscan[tid] - c;   // segment start in gt

    float lp = 0.0f, la = 0.0f, lw = 0.0f, ls = 0.0f;
    if (c > 0 && i < n) {
        const float* pp = pred + (size_t)i * 6;
        float px = pp[0], py = pp[1], p2 = pp[2];
        float p3 = pp[3], p4 = pp[4], p5 = pp[5];

        const float* gp = gt + (size_t)o * 6;
        __builtin_prefetch(gp, 0, 0);            // global_prefetch_b8

        float bestD = INFINITY;
        float bg2 = 0.f, bg3 = 0.f, bg4 = 0.f, bg5 = 0.f;
        for (int k = 0; k < c; ++k) {
            float g0 = gp[0], g1 = gp[1], g2 = gp[2];
            float g3 = gp[3], g4 = gp[4], g5 = gp[5];
            float dx = px - g0, dy = py - g1;
            float sqd = dx * dx + dy * dy;
            if (sqd < bestD) {                   // strict < => first-index tie-break
                bestD = sqd; bg2 = g2; bg3 = g3; bg4 = g4; bg5 = g5;
            }
            gp += 6;
        }

        // loss_pos: 1 - max(exp(-d/k)) == 1 - exp(-min(d)/k)
        lp = 1.0f - expf(-bestD / 0.045000000000000005f);   // 2*SIGMA^2
        la = fabsf(p2 - bg2) + fabsf(p3 - bg3);
        float d  = p4 - bg4;
        float ad = fabsf(d);
        lw = (ad < 1.0f) ? (0.5f * d * d) : (ad - 0.5f);
        float x = p5, t = bg5;
        float bce = fmaxf(x, 0.0f) - x * t + log1pf(expf(-fabsf(x)));
        ls = (t > 0.0f) ? bce : 0.0f;
    }

    sred[0][tid] = lp; sred[1][tid] = la; sred[2][tid] = lw; sred[3][tid] = ls;
    __syncthreads();
    for (int st = BS / 2; st > 0; st >>= 1) {
        if (tid < st) {
            sred[0][tid] += sred[0][tid + st];
            sred[1][tid] += sred[1][tid + st];
            sred[2][tid] += sred[2][tid + st];
            sred[3][tid] += sred[3][tid + st];
        }
        __syncthreads();
    }
    if (tid == 0) {
        float4 v = make_float4(sred[0][0], sred[1][0], sred[2][0], sred[3][0]);
        ((float4*)partials)[blockIdx.x] = v;
    }
}

// ---- Kernel 4: WMMA-based reduction of partials + final output -------------
// A = ones(16x4) f32  =>  D[i][j] = sum_K B[K][j] + C[i][j]  (K=4 per issue).
// Column j<4 of B carries loss component j from 4 partial rows per issue.
// Single wave of 32 lanes; no divergence (masked loads via clamped indices)
// so EXEC is all-ones at every v_wmma issue.
__global__ void k_finalize(const float* __restrict__ partials,
                           float* __restrict__ out, int nb, int n)
{
    __shared__ float s4[4];
    int lane = threadIdx.x;                 // 0..31
    int j    = lane & 15;                   // column within B
    float mask = (j < 4) ? 1.0f : 0.0f;
    int   jj   = (j < 4) ? j : 0;
    int   half = (lane >> 4) * 2;           // lanes 16..31 carry rows +2,+3

    v2f a; a.x = 1.0f; a.y = 1.0f;          // all-ones A matrix
    v8f acc = {};

    for (int base = 0; base < nb; base += 4) {
        int r0 = base + half;
        int r1 = r0 + 1;
        float m0 = (r0 < nb) ? mask : 0.0f;
        float m1 = (r1 < nb) ? mask : 0.0f;
        int c0 = (r0 < nb) ? r0 : 0;
        int c1 = (r1 < nb) ? r1 : 0;
        v2f b;
        b.x = partials[(size_t)c0 * 4 + jj] * m0;
        b.y = partials[(size_t)c1 * 4 + jj] * m1;
        acc = __builtin_amdgcn_wmma_f32_16x16x4_f32(
                  false, a, false, b, (short)0, acc, false, false);
    }

    // D row M=0 lives in VGPR0 (acc[0]), N = lane for lanes 0..15
    if (lane < 4) s4[lane] = acc[0];
    __syncthreads();
    if (lane == 0) {
        float fn = (float)n;
        float lp = s4[0] / fn;
        float la = s4[1] / fn;
        float lw = s4[2] / fn;
        float ls = s4[3] / fn;
        out[0] = lp; out[1] = la; out[2] = lw; out[3] = ls;
        out[4] = lp + la + lw + 0.5f * ls;   // POS/ANG/WID=1.0, SCORE=0.5
    }
}

// ---------------------------------------------------------------------------
extern "C" void kernel_launch(void* const* d_in, const int* in_sizes, int n_in,
                              void* d_out, int out_size, void* d_ws, size_t ws_size,
                              hipStream_t stream)
{
    const float* pred   = (const float*)d_in[0];   // (N,6) f32
    const float* gt     = (const float*)d_in[1];   // (M,6) f32
    const int*   counts = (const int*)d_in[2];     // (N,)  i32

    int n    = in_sizes[2];                 // N = 262144
    int nblk = (n + BS - 1) / BS;           // 1024 (<= SCAN)

    int*   blockSums    = (int*)d_ws;
    int*   blockOffsets = blockSums + SCAN;
    float* partials     = (float*)(blockOffsets + SCAN);

    k_block_sums <<<nblk, BS,   0, stream>>>(counts, blockSums, n);
    k_scan_blocks<<<1,    SCAN, 0, stream>>>(blockSums, blockOffsets, nblk);
    k_main       <<<nblk, BS,   0, stream>>>(pred, gt, counts, blockOffsets,
                                             partials, n);
    k_finalize   <<<1,    32,   0, stream>>>(partials, (float*)d_out, nblk, n);
}